// ELBO_27023934226450
// MI455X (gfx1250) — compile-verified
//
#include <hip/hip_runtime.h>

typedef float v2f __attribute__((ext_vector_type(2)));
typedef float v8f __attribute__((ext_vector_type(8)));
typedef int   v2i __attribute__((ext_vector_type(2)));

#define B_N     4096
#define CTX_N   20
#define EMBED_N 100
#define VOCAB_N 50000

// One wave (32 lanes) produces 16 output rows.
// A-matrix layout for V_WMMA_F32_16X16X4_F32 (16x4 f32):
//   lanes 0-15 : M=lane,    VGPR0=K0, VGPR1=K1
//   lanes 16-31: M=lane-16, VGPR0=K2, VGPR1=K3
// B = all-ones (4x16) -> every column of D is the 4-wide row sum; accumulate C
// across 25 embed tiles + 5 context tiles.
__global__ __launch_bounds__(32) void elbo_wmma_kernel(
    const int*   __restrict__ context,       // [B, CTX]
    const float* __restrict__ mu_lambda,     // [B, EMBED]
    const float* __restrict__ sigma_lambda,  // [B, EMBED]
    const float* __restrict__ mu_x,          // [B, EMBED]
    const float* __restrict__ sigma_x,       // [B, EMBED]
    const float* __restrict__ decoded,       // [B, VOCAB]
    float*       __restrict__ out)           // [B]
{
    const int lane    = threadIdx.x & 31;
    const int rowBase = blockIdx.x * 16;
    const int m       = lane & 15;           // row within the 16-row tile
    const int kp      = (lane >> 4) << 1;    // lanes<16 -> K{0,1}; lanes>=16 -> K{2,3}
    const int row     = rowBase + m;

    v8f c = {0.f, 0.f, 0.f, 0.f, 0.f, 0.f, 0.f, 0.f};
    const v2f bones = {1.0f, 1.0f};          // all-ones B matrix (layout-invariant)

    // ---- KL terms over EMBED: 25 tiles of K=4 ----
    const float* ml = mu_lambda    + (size_t)row * EMBED_N;
    const float* sl = sigma_lambda + (size_t)row * EMBED_N;
    const float* mx = mu_x         + (size_t)row * EMBED_N;
    const float* sx = sigma_x      + (size_t)row * EMBED_N;

    for (int k0 = 0; k0 < EMBED_N; k0 += 4) {
        const v2f vml = *(const v2f*)(ml + k0 + kp);
        const v2f vsl = *(const v2f*)(sl + k0 + kp);
        const v2f vmx = *(const v2f*)(mx + k0 + kp);
        const v2f vsx = *(const v2f*)(sx + k0 + kp);
        v2f a;
#pragma unroll
        for (int j = 0; j < 2; ++j) {
            const float d  = vml[j] - vmx[j];
            const float s2 = vsx[j] * vsx[j];
            a[j] = logf(vsx[j] / vsl[j])
                 + (vsl[j] * vsl[j] + d * d) / (2.0f * s2)
                 - 0.5f;
        }
        c = __builtin_amdgcn_wmma_f32_16x16x4_f32(
                false, a, false, bones, (short)0, c, false, false);
    }

    // ---- -log-likelihood terms over CTX: 5 tiles of K=4 ----
    const int*   ctx = context + (size_t)row * CTX_N;
    const float* dec = decoded + (size_t)row * VOCAB_N;

    for (int k0 = 0; k0 < CTX_N; k0 += 4) {
        const v2i idx = *(const v2i*)(ctx + k0 + kp);
        v2f a;
#pragma unroll
        for (int j = 0; j < 2; ++j) {
            const float g = __builtin_nontemporal_load(dec + idx[j]);
            a[j] = -logf(g);                 // negative ELBO uses -ll
        }
        c = __builtin_amdgcn_wmma_f32_16x16x4_f32(
                false, a, false, bones, (short)0, c, false, false);
    }

    // ---- extract: D[j][n] identical for all columns n ----
    // lanes 0-15 hold rows rowBase+0..7 in c[0..7]; lanes 16-31 hold rows rowBase+8..15
    const int j = lane & 7;
    float val = c[0];
#pragma unroll
    for (int t = 1; t < 8; ++t)
        if (j == t) val = c[t];

    if (lane < 8) {
        out[rowBase + lane] = val;
    } else if (lane >= 16 && lane < 24) {
        out[rowBase + 8 + (lane - 16)] = val;
    }
}

extern "C" void kernel_launch(void* const* d_in, const int* in_sizes, int n_in,
                              void* d_out, int out_size, void* d_ws, size_t ws_size,
                              hipStream_t stream) {
    const int*   context      = (const int*)  d_in[0];
    const float* mu_lambda    = (const float*)d_in[1];
    const float* sigma_lambda = (const float*)d_in[2];
    const float* mu_x         = (const float*)d_in[3];
    const float* sigma_x      = (const float*)d_in[4];
    const float* decoded      = (const float*)d_in[5];
    float*       out          = (float*)d_out;

    dim3 grid(B_N / 16);   // 256 single-wave workgroups, 16 rows each
    dim3 block(32);        // wave32: one wave per block, no barrier resources
    elbo_wmma_kernel<<<grid, block, 0, stream>>>(
        context, mu_lambda, sigma_lambda, mu_x, sigma_x, decoded, out);
}